// GeneralisedShapeletTransform_77429670412411
// MI455X (gfx1250) — compile-verified
//
#include <hip/hip_runtime.h>
#include <math.h>

// ---------------------------------------------------------------------------
// Generalised shapelet transform, MI455X (gfx1250).
//
// Shapes: T=512, B=32, C=8, K=32, m=32, S=256.  Output (B,K) fp32.
//
// Key identity: sum_j (a^2 + a b + b^2) over segments equals d^T M d with
// M tridiagonal: diag = [1,2,...,2,1], offdiag = 1/2.  We fold M into the
// WMMA B operand, so C = D x (M D)^T and diag(C)[b] = d_b^T M d_b, computed
// entirely in fp32 via V_WMMA_F32_16X16X4_F32 (matches reference precision).
// ---------------------------------------------------------------------------

typedef __attribute__((ext_vector_type(2))) float v2f;
typedef __attribute__((ext_vector_type(4))) float v4f;
typedef __attribute__((ext_vector_type(8))) float v8f;

#define T_KNOTS 512
#define BATCH   32
#define CH      8
#define KSH     32
#define MKNOT   32
#define SSAMP   256
#define RLEN    (MKNOT * CH)      /* 256 = reduction length (j,c) */
#define RPADROW 8                 /* one j (8 channels) of zero padding each side */

__global__ void shapelet_init_out(int* __restrict__ outBits) {
    int i = blockIdx.x * blockDim.x + threadIdx.x;
    if (i < BATCH * KSH) outBits[i] = 0x7f800000;   // +inf
}

__global__ __launch_bounds__(32)
void shapelet_wmma_kernel(const float* __restrict__ times,
                          const float* __restrict__ path,
                          const float* __restrict__ lengths,
                          const float* __restrict__ shapelets,
                          int* __restrict__ outBits)
{
    // dpad rows: [0..7] zero pad, [8..263] = r = j*8+c, [264..271] zero pad.
    __shared__ float dpad[(RPADROW + RLEN + RPADROW) * 32];
    __shared__ float wA[MKNOT];
    __shared__ int   idxA[MKNOT];
    __shared__ float ctile[2][8][32];

    const int lane = threadIdx.x;          // wave32: lane == thread
    const int s    = blockIdx.x;           // start-position sample
    const int k    = blockIdx.y;           // shapelet index

    const float t0  = times[0];
    const float tE  = times[T_KNOTS - 1];
    const float len = lengths[k];
    const float start = t0 + (float)s * (1.0f / (SSAMP - 1)) * (tE - t0 - len);

    // Zero the padding rows (512 floats total).
    for (int i = lane; i < RPADROW * 32; i += 32) {
        dpad[i] = 0.0f;
        dpad[(RPADROW + RLEN) * 32 + i] = 0.0f;
    }

    // ---- Phase 1: lane j -> sample position, knot index, lerp weight ------
    {
        const int j = lane;
        float u = (float)j * (1.0f / (MKNOT - 1)) * len;
        float q = start + u;
        // times are uniform: searchsorted(right)-1 == floor((q-t0)*(T-1)/(tE-t0))
        float x = (q - t0) * ((float)(T_KNOTS - 1) / (tE - t0));
        int idx = (int)floorf(x);
        idx = idx < 0 ? 0 : (idx > T_KNOTS - 2 ? T_KNOTS - 2 : idx);
        wA[j]   = x - (float)idx;
        idxA[j] = idx;
    }
    __syncthreads();

    // ---- Phase 2: lane b -> build D(b, r=(j,c)) = lerp(path) - shapelet ---
    const int b = lane;
    const float* __restrict__ pb  = path + (size_t)b * T_KNOTS * CH;
    const float* __restrict__ shk = shapelets + (size_t)k * MKNOT * CH;
    __builtin_prefetch(pb, 0, 1);          // global_prefetch_b8: warm path row

    for (int j = 0; j < MKNOT; ++j) {
        const int   idx = idxA[j];
        const float w   = wA[j];
        const float* pl = pb + idx * CH;
        v4f pl0 = *(const v4f*)(pl);
        v4f pl1 = *(const v4f*)(pl + 4);
        v4f pr0 = *(const v4f*)(pl + CH);
        v4f pr1 = *(const v4f*)(pl + CH + 4);
        v4f s0  = *(const v4f*)(shk + j * CH);      // uniform -> scalar loads
        v4f s1  = *(const v4f*)(shk + j * CH + 4);
        v4f d0  = pl0 + w * (pr0 - pl0) - s0;
        v4f d1  = pl1 + w * (pr1 - pl1) - s1;
        const int rbase = RPADROW + j * CH;
#pragma unroll
        for (int c = 0; c < 4; ++c) {
            dpad[(rbase + c) * 32 + b]     = d0[c];
            dpad[(rbase + 4 + c) * 32 + b] = d1[c];
        }
    }
    __syncthreads();

    // ---- Phase 3: WMMA chains.  acc_t = D_t x (M D_t)^T, t = batch tile --
    v8f acc0 = {};
    v8f acc1 = {};
    const int half = lane >> 4;            // A/B fragment K-half select
    const int li   = lane & 15;
    const int kOff = half * 2;             // VGPR0 holds K=0 (lanes 0-15) / K=2 (16-31)

    for (int rr = 0; rr < RLEN / 4; ++rr) {
        const int r0 = rr * 4 + kOff;      // absolute r for .x ; .y is r0+1
        const int rowX = (RPADROW + r0) * 32;
        const int rowY = rowX + 32;
        // alpha(r): 1 at j==0 or j==m-1, else 2
        const float alx = (r0 < CH || r0 >= RLEN - CH) ? 1.0f : 2.0f;
        const float aly = (r0 + 1 < CH || r0 + 1 >= RLEN - CH) ? 1.0f : 2.0f;
#pragma unroll
        for (int t = 0; t < 2; ++t) {
            const int col = t * 16 + li;   // A row (b) == B col (b') for diag use
            float ax = dpad[rowX + col];
            float ay = dpad[rowY + col];
            float xm = dpad[rowX - CH * 32 + col];   // d(j-1) neighbors
            float xp = dpad[rowX + CH * 32 + col];   // d(j+1)
            float ym = dpad[rowY - CH * 32 + col];
            float yp = dpad[rowY + CH * 32 + col];
            v2f A;  A.x = ax;  A.y = ay;
            v2f Bv; Bv.x = alx * ax + 0.5f * (xm + xp);   // (M d) row r0
            Bv.y = aly * ay + 0.5f * (ym + yp);           // (M d) row r0+1
            if (t == 0)
                acc0 = __builtin_amdgcn_wmma_f32_16x16x4_f32(
                    false, A, false, Bv, (short)0, acc0, false, false);
            else
                acc1 = __builtin_amdgcn_wmma_f32_16x16x4_f32(
                    false, A, false, Bv, (short)0, acc1, false, false);
        }
    }

    // ---- Phase 4: pull diagonal, scale, sqrt, atomic min over s ----------
#pragma unroll
    for (int v = 0; v < 8; ++v) {
        ctile[0][v][lane] = acc0[v];
        ctile[1][v][lane] = acc1[v];
    }
    __syncthreads();

    // C/D layout: VGPR v, lanes 0-15 -> (M=v, N=lane); lanes 16-31 -> (M=v+8, N=lane-16)
    const int bt = lane >> 4;
    const int i  = lane & 15;
    const int vv = (i < 8) ? i : (i - 8);
    const int cl = (i < 8) ? i : (i + 16);
    const float quad = ctile[bt][vv][cl];            // d_b^T M d_b

    const float dt   = len * (1.0f / (MKNOT - 1));
    const float disc = sqrtf(quad * (dt * (1.0f / 3.0f)) + 1e-12f);

    // disc >= 0: float order == int-bit order; out[b*K + k]
    atomicMin(outBits + lane * KSH + k, __float_as_int(disc));
}

extern "C" void kernel_launch(void* const* d_in, const int* in_sizes, int n_in,
                              void* d_out, int out_size, void* d_ws, size_t ws_size,
                              hipStream_t stream) {
    const float* times     = (const float*)d_in[0];
    const float* path      = (const float*)d_in[1];
    const float* lengths   = (const float*)d_in[2];
    const float* shapelets = (const float*)d_in[3];
    int* outBits = (int*)d_out;

    // Re-initialize output to +inf every call (deterministic; harness poisons).
    shapelet_init_out<<<dim3((BATCH * KSH + 255) / 256), dim3(256), 0, stream>>>(outBits);

    // One wave32 per (start, shapelet) job: 256 x 32 = 8192 independent waves.
    shapelet_wmma_kernel<<<dim3(SSAMP, KSH), dim3(32), 0, stream>>>(
        times, path, lengths, shapelets, outBits);
}